// GCNBlock_56547539419677
// MI455X (gfx1250) — compile-verified
//
#include <hip/hip_runtime.h>
#include <hip/hip_bf16.h>

typedef __attribute__((ext_vector_type(2))) float v2f;
typedef __attribute__((ext_vector_type(8))) float v8f;
typedef int v4i __attribute__((vector_size(4 * sizeof(int))));

#define FEAT 128
#define XS_PITCH 132         // padded LDS pitch for the x tile (bank-conflict avoidance)
#define ROWS_PER_BLOCK 64    // 4 row sub-tiles of 16; amortizes W staging, reuses B frags

#if defined(__HIP_DEVICE_COMPILE__) && __has_builtin(__builtin_amdgcn_global_load_async_to_lds_b128)
#define USE_ASYNC_LDS 1
#endif

#ifdef USE_ASYNC_LDS
__device__ __forceinline__ __attribute__((address_space(1))) v4i*
to_global_v4(const void* p) {
    return (__attribute__((address_space(1))) v4i*)(uintptr_t)p;
}
__device__ __forceinline__ __attribute__((address_space(3))) v4i*
to_lds_v4(void* p) {
    // as(3) value == low 32 bits of the generic shared pointer (aperture mapping)
    return (__attribute__((address_space(3))) v4i*)(unsigned)(uintptr_t)p;
}
__device__ __forceinline__ void wait_asynccnt0() {
#if __has_builtin(__builtin_amdgcn_s_wait_asynccnt)
    __builtin_amdgcn_s_wait_asynccnt(0);
#else
    asm volatile("s_wait_asynccnt 0" ::: "memory");
#endif
}
#endif

// ---------------------------------------------------------------- utilities
__global__ void fill_f32_kernel(float* __restrict__ p, float v, int n) {
    int i = blockIdx.x * blockDim.x + threadIdx.x;
    if (i < n) p[i] = v;
}

// deg[col[e]] += ew[e]   (deg pre-initialized to 1.0 for self loops)
__global__ void deg_accum_kernel(const int* __restrict__ col,
                                 const float* __restrict__ ew,
                                 float* __restrict__ deg, int n_edges) {
    int e = blockIdx.x * blockDim.x + threadIdx.x;
    if (e < n_edges) atomicAdd(&deg[col[e]], ew[e]);
}

// dinv = deg>0 ? rsqrt(deg) : 0 ; selfnorm = dinv*dinv
__global__ void dinv_kernel(const float* __restrict__ deg,
                            float* __restrict__ dinv,
                            float* __restrict__ selfnorm, int n_nodes) {
    int i = blockIdx.x * blockDim.x + threadIdx.x;
    if (i < n_nodes) {
        float d = deg[i];
        float r = (d > 0.0f) ? __frsqrt_rn(d) : 0.0f;
        dinv[i] = r;
        selfnorm[i] = r * r;
    }
}

// norm[e] = dinv[row]*ew*dinv[col]
__global__ void edge_norm_kernel(const int* __restrict__ row,
                                 const int* __restrict__ col,
                                 const float* __restrict__ ew,
                                 const float* __restrict__ dinv,
                                 float* __restrict__ norm, int n_edges) {
    int e = blockIdx.x * blockDim.x + threadIdx.x;
    if (e < n_edges) norm[e] = dinv[row[e]] * ew[e] * dinv[col[e]];
}

// cnt[batch[i]] += 1
__global__ void count_kernel(const int* __restrict__ batch,
                             float* __restrict__ cnt, int n_nodes) {
    int i = blockIdx.x * blockDim.x + threadIdx.x;
    if (i < n_nodes) atomicAdd(&cnt[batch[i]], 1.0f);
}

// ------------------------------------------------------------- WMMA GEMM
// Y[n, 128] = X[n, 128] @ W[128, 128]
// One block = 64 output rows x 128 cols. 8 waves; wave w owns a 16-col tile
// across 4 row sub-tiles (4 accumulators, B fragment reused 4x per k-step).
// Uses V_WMMA_F32_16X16X4_F32, K stepped by 4.
__global__ __launch_bounds__(256) void gemm_wmma_kernel(
        const float* __restrict__ X, const float* __restrict__ W,
        float* __restrict__ Y, int n_rows) {
    __shared__ float ws[FEAT * FEAT];               // 64 KB: full W
    __shared__ float xs[ROWS_PER_BLOCK * XS_PITCH]; // 33 KB: padded x tile

    const int tid = threadIdx.x;
    const int rowbase = blockIdx.x * ROWS_PER_BLOCK;

    // ---- Stage W (4096 float4) and the x tile (64 rows x 32 float4) ----
#ifdef USE_ASYNC_LDS
    for (int i = tid; i < (FEAT * FEAT) / 4; i += 256) {
        __builtin_amdgcn_global_load_async_to_lds_b128(
            to_global_v4((const float4*)W + i), to_lds_v4((float4*)ws + i), 0, 0);
    }
    for (int i = tid; i < ROWS_PER_BLOCK * 32; i += 256) {
        int r = i >> 5, c4 = i & 31;
        int gr = rowbase + r;
        float4* dst = (float4*)&xs[r * XS_PITCH + c4 * 4];
        if (gr < n_rows) {
            __builtin_amdgcn_global_load_async_to_lds_b128(
                to_global_v4((const float4*)X + (size_t)gr * 32 + c4), to_lds_v4(dst), 0, 0);
        } else {
            *dst = make_float4(0.f, 0.f, 0.f, 0.f);   // disjoint addrs; ds path
        }
    }
    wait_asynccnt0();
#else
    {
        const float4* W4 = (const float4*)W;
        float4* ws4 = (float4*)ws;
        for (int i = tid; i < (FEAT * FEAT) / 4; i += 256) ws4[i] = W4[i];
        for (int i = tid; i < ROWS_PER_BLOCK * 32; i += 256) {
            int r = i >> 5, c4 = i & 31;
            int gr = rowbase + r;
            float4 v = make_float4(0.f, 0.f, 0.f, 0.f);
            if (gr < n_rows) v = ((const float4*)X)[(size_t)gr * 32 + c4];
            *(float4*)&xs[r * XS_PITCH + c4 * 4] = v;
        }
    }
#endif
    __syncthreads();

    const int wave = tid >> 5;        // 0..7 -> column tile
    const int lane = tid & 31;
    const int half = lane >> 4;       // lane group 0/1
    const int ln   = lane & 15;
    const int colbase = wave * 16;

    v8f c0 = {}, c1 = {}, c2 = {}, c3 = {};
    #pragma unroll 4
    for (int k0 = 0; k0 < FEAT; k0 += 4) {
        // B 4x16 fragment: row-striped across lanes; lane half selects K rows 2h/2h+1
        v2f b;
        b.x = ws[(k0 + 2 * half) * FEAT + colbase + ln];
        b.y = ws[(k0 + 2 * half + 1) * FEAT + colbase + ln];
        // A 16x4 fragments for the 4 row sub-tiles (ISA 7.12.2 layout)
        v2f a0, a1, a2, a3;
        const int kk = k0 + 2 * half;
        a0.x = xs[(ln +  0) * XS_PITCH + kk];  a0.y = xs[(ln +  0) * XS_PITCH + kk + 1];
        a1.x = xs[(ln + 16) * XS_PITCH + kk];  a1.y = xs[(ln + 16) * XS_PITCH + kk + 1];
        a2.x = xs[(ln + 32) * XS_PITCH + kk];  a2.y = xs[(ln + 32) * XS_PITCH + kk + 1];
        a3.x = xs[(ln + 48) * XS_PITCH + kk];  a3.y = xs[(ln + 48) * XS_PITCH + kk + 1];
        c0 = __builtin_amdgcn_wmma_f32_16x16x4_f32(false, a0, false, b, (short)0, c0, false, false);
        c1 = __builtin_amdgcn_wmma_f32_16x16x4_f32(false, a1, false, b, (short)0, c1, false, false);
        c2 = __builtin_amdgcn_wmma_f32_16x16x4_f32(false, a2, false, b, (short)0, c2, false, false);
        c3 = __builtin_amdgcn_wmma_f32_16x16x4_f32(false, a3, false, b, (short)0, c3, false, false);
    }

    // D 16x16 f32: VGPR r -> M = r + 8*half, N = ln
    v8f acc[4] = { c0, c1, c2, c3 };
    #pragma unroll
    for (int rt = 0; rt < 4; ++rt) {
        #pragma unroll
        for (int r = 0; r < 8; ++r) {
            int gr = rowbase + rt * 16 + r + 8 * half;
            if (gr < n_rows) Y[(size_t)gr * FEAT + colbase + ln] = acc[rt][r];
        }
    }
}

// agg[i*128 + j] = b[j]
__global__ void bias_init_kernel(float* __restrict__ agg,
                                 const float* __restrict__ b, int n_elems) {
    int i = blockIdx.x * blockDim.x + threadIdx.x;
    if (i < n_elems) agg[i] = b[i & (FEAT - 1)];
}

// One wave per edge: agg[col[e], :] += norm[e] * xw[row[e], :]
__global__ __launch_bounds__(256) void edge_scatter_kernel(
        const int* __restrict__ row, const int* __restrict__ col,
        const float* __restrict__ norm, const float* __restrict__ xw,
        float* __restrict__ agg, int n_edges) {
    int e = blockIdx.x * 8 + (threadIdx.x >> 5);
    if (e >= n_edges) return;
    int lane = threadIdx.x & 31;
    if (lane == 0 && e + 4096 < n_edges) {
        __builtin_prefetch(&norm[e + 4096], 0, 0);   // global_prefetch_b8
    }
    float w = norm[e];
    int r = row[e], c = col[e];
    float4 v = ((const float4*)(xw + (size_t)r * FEAT))[lane];
    float* dst = agg + (size_t)c * FEAT + lane * 4;
    atomicAdd(dst + 0, w * v.x);
    atomicAdd(dst + 1, w * v.y);
    atomicAdd(dst + 2, w * v.z);
    atomicAdd(dst + 3, w * v.w);
}

// agg += selfnorm[node] * xw   (runs after all edge atomics; no races)
__global__ void self_add_kernel(float* __restrict__ agg,
                                const float* __restrict__ xw,
                                const float* __restrict__ selfnorm, int n_elems) {
    int i = blockIdx.x * blockDim.x + threadIdx.x;
    if (i < n_elems) agg[i] += selfnorm[i >> 7] * xw[i];
}

// optional in-place relu, then embsum[batch[node]*128 + j] += agg
__global__ void relu_pool_kernel(float* __restrict__ agg,
                                 const int* __restrict__ batch,
                                 float* __restrict__ embsum,
                                 int n_elems, int do_relu) {
    int i = blockIdx.x * blockDim.x + threadIdx.x;
    if (i >= n_elems) return;
    float v = agg[i];
    if (do_relu) { v = fmaxf(v, 0.0f); agg[i] = v; }
    atomicAdd(&embsum[((size_t)batch[i >> 7] << 7) + (i & (FEAT - 1))], v);
}

// out = embsum / max(cnt, 1)
__global__ void pool_norm_kernel(const float* __restrict__ embsum,
                                 const float* __restrict__ cnt,
                                 float* __restrict__ out, int n_elems) {
    int i = blockIdx.x * blockDim.x + threadIdx.x;
    if (i < n_elems) out[i] = embsum[i] / fmaxf(cnt[i >> 7], 1.0f);
}

// ---------------------------------------------------------------- launcher
extern "C" void kernel_launch(void* const* d_in, const int* in_sizes, int n_in,
                              void* d_out, int out_size, void* d_ws, size_t ws_size,
                              hipStream_t stream) {
    const float* x  = (const float*)d_in[0];
    const int*   ei = (const int*)d_in[1];
    const float* ew = (const float*)d_in[2];
    const int*   batch = (const int*)d_in[3];
    const float* Wl[3] = { (const float*)d_in[4], (const float*)d_in[6], (const float*)d_in[8] };
    const float* bl[3] = { (const float*)d_in[5], (const float*)d_in[7], (const float*)d_in[9] };

    const int n_nodes = in_sizes[0] / FEAT;      // 100000
    const int n_edges = in_sizes[2];             // 1600000
    const int n_elems = n_nodes * FEAT;          // 12.8M
    const int n_graphs = 128;

    const int* erow = ei;                        // edge_index[0]
    const int* ecol = ei + n_edges;              // edge_index[1]

    // workspace layout (floats)
    float* ws = (float*)d_ws;
    size_t off = 0;
    auto take = [&](size_t n) { float* p = ws + off; off += (n + 255) & ~(size_t)255; return p; };
    float* deg      = take(n_nodes);
    float* dinv     = take(n_nodes);
    float* selfnorm = take(n_nodes);
    float* norm     = take(n_edges);
    float* cnt      = take(n_graphs);
    float* embsum   = take(n_graphs * FEAT);
    float* xw       = take(n_elems);
    float* hA       = take(n_elems);
    float* hB       = take(n_elems);

    float* out = (float*)d_out;
    float* emb_out = out + in_sizes[0];          // after the x copy

    const int T = 256;
    auto cdiv = [](int a, int b) { return (a + b - 1) / b; };

    // 0) outputs[0] = x (pass-through)
    (void)hipMemcpyAsync(out, x, (size_t)in_sizes[0] * sizeof(float),
                         hipMemcpyDeviceToDevice, stream);

    // 1) normalization precompute
    fill_f32_kernel<<<cdiv(n_nodes, T), T, 0, stream>>>(deg, 1.0f, n_nodes);   // self-loop weight
    fill_f32_kernel<<<1, T, 0, stream>>>(cnt, 0.0f, n_graphs);
    deg_accum_kernel<<<cdiv(n_edges, T), T, 0, stream>>>(ecol, ew, deg, n_edges);
    dinv_kernel<<<cdiv(n_nodes, T), T, 0, stream>>>(deg, dinv, selfnorm, n_nodes);
    edge_norm_kernel<<<cdiv(n_edges, T), T, 0, stream>>>(erow, ecol, ew, dinv, norm, n_edges);
    count_kernel<<<cdiv(n_nodes, T), T, 0, stream>>>(batch, cnt, n_nodes);

    // 2) three GCN layers
    const float* h = x;
    float* aggs[3] = { hA, hB, hA };   // hA free again by layer 2
    for (int layer = 0; layer < 3; ++layer) {
        float* agg = aggs[layer];
        gemm_wmma_kernel<<<cdiv(n_nodes, ROWS_PER_BLOCK), T, 0, stream>>>(h, Wl[layer], xw, n_nodes);
        bias_init_kernel<<<cdiv(n_elems, T), T, 0, stream>>>(agg, bl[layer], n_elems);
        edge_scatter_kernel<<<cdiv(n_edges, 8), T, 0, stream>>>(erow, ecol, norm, xw, agg, n_edges);
        self_add_kernel<<<cdiv(n_elems, T), T, 0, stream>>>(agg, xw, selfnorm, n_elems);
        fill_f32_kernel<<<cdiv(n_graphs * FEAT, T), T, 0, stream>>>(embsum, 0.0f, n_graphs * FEAT);
        relu_pool_kernel<<<cdiv(n_elems, T), T, 0, stream>>>(agg, batch, embsum, n_elems,
                                                             (layer < 2) ? 1 : 0);
        pool_norm_kernel<<<cdiv(n_graphs * FEAT, T), T, 0, stream>>>(
            embsum, cnt, emb_out + (size_t)layer * n_graphs * FEAT, n_graphs * FEAT);
        h = agg;
    }
}